// MixSample_40973988004171
// MI455X (gfx1250) — compile-verified
//
#include <hip/hip_runtime.h>
#include <hip/hip_bf16.h>
#include <math.h>
#include <stdint.h>

// ---------------------------------------------------------------------------
// MixSample fused pipeline for gfx1250 (MI455X), all-f32 via V_WMMA_F32_16X16X4_F32
// Stage1 streams x frames with async global->LDS (ASYNCcnt) double buffering.
// Shapes: N=16, C=64, T=30, H=64, W=44, OUT_H=16, OUT_W=20, aout_h=16 (no pad)
// ---------------------------------------------------------------------------

typedef __attribute__((ext_vector_type(2))) float v2f;
typedef __attribute__((ext_vector_type(8))) float v8f;
typedef __attribute__((ext_vector_type(4))) int   v4i;
typedef __attribute__((address_space(1))) v4i     as1_v4i;
typedef __attribute__((address_space(3))) v4i     as3_v4i;

#define NB    16
#define CC    64
#define TT    30
#define HH    64
#define WW    44
#define WP    48      // W padded for WMMA K/N tiling
#define QP    32      // OUT_W (20) padded
#define OUTW  20
#define OUTH  16
#define HWSZ  (OUTH*OUTW)          // 320
#define OUT_MAIN ((size_t)NB*CC*TT*HWSZ)   // 9,830,400

// workspace offsets (in floats)
#define OFF_FXT 0            // [N][48][32]  = 24576
#define OFF_FY  24576        // [N][16][64]  = 16384
#define OFF_GAM 40960        // [N]
#define OFF_TW0 40976        // [N][30]
#define OFF_TW1 41456        // [N][30]
#define OFF_TI0 41936        // [N][30] (int)
#define OFF_TI1 42416        // [N][30] (int)
#define OFF_G   43008        // [N][C][T][16][20] = 9,830,400

__device__ __forceinline__ v8f wmma_f32(v2f a, v2f b, v8f c) {
  // D(16x16) = A(16x4) * B(4x16) + C   -- f32 in / f32 acc
  return __builtin_amdgcn_wmma_f32_16x16x4_f32(false, a, false, b, (short)0, c,
                                               false, false);
}

// ---------------------------------------------------------------------------
// Stage 0: MLP -> attention params -> filterbanks + temporal tables + params out
// ---------------------------------------------------------------------------
__global__ void __launch_bounds__(256)
prep_kernel(const float* __restrict__ param_x, const float* __restrict__ W1,
            const float* __restrict__ b1, const float* __restrict__ W2,
            const float* __restrict__ b2, float* __restrict__ ws,
            float* __restrict__ params_out)
{
  __shared__ float hid[NB][32];
  __shared__ float pv[NB][8];
  __shared__ float prm[NB][8];   // 0:dt 1:dx 2:dy 3:sigma2 4:delta 5:delta_t 6:gamma

  const int tid = threadIdx.x;
  float* FxT = ws + OFF_FXT;
  float* Fy  = ws + OFF_FY;
  float* gam = ws + OFF_GAM;
  float* tw0 = ws + OFF_TW0;
  float* tw1 = ws + OFF_TW1;
  int*   ti0 = (int*)(ws + OFF_TI0);
  int*   ti1 = (int*)(ws + OFF_TI1);

  // hid = tanh(param_x @ W1 + b1)   (16x64 @ 64x32)
  for (int i = tid; i < NB * 32; i += 256) {
    int n = i >> 5, j = i & 31;
    float acc = b1[j];
    for (int k = 0; k < 64; ++k) acc += param_x[n * 64 + k] * W1[k * 32 + j];
    hid[n][j] = tanhf(acc);
  }
  __syncthreads();
  // p = hid @ W2 + b2   (16x32 @ 32x7)
  for (int i = tid; i < NB * 7; i += 256) {
    int n = i / 7, j = i % 7;
    float acc = b2[j];
    for (int k = 0; k < 32; ++k) acc += hid[n][k] * W2[k * 7 + j];
    pv[n][j] = acc;
  }
  __syncthreads();
  if (tid < NB) {
    int n = tid;
    float dt      = tanhf(pv[n][0]) * 6.0f  + 15.0f;  // atten_in_t/2 + anchor_t
    float dx      = tanhf(pv[n][1]) * 20.0f + 22.0f;  // atten_in_w/2 + anchor_x
    float dy      = tanhf(pv[n][2]) * 16.0f + 24.0f;  // atten_in_h/2 + anchor_y
    float sigma2  = __expf(pv[n][3]);
    float delta_t = __expf(pv[n][4]) * 0.4f;          // T_RATE
    float delta   = __expf(pv[n][5]) * (10.0f / 11.0f); // W_RATE
    float gamma   = 1.0f / (1.0f + __expf(-pv[n][6]));
    prm[n][0] = dt; prm[n][1] = dx; prm[n][2] = dy; prm[n][3] = sigma2;
    prm[n][4] = delta; prm[n][5] = delta_t; prm[n][6] = gamma;
    gam[n] = gamma;
    // params tail of d_out: stack order [dt,dx,dy,sigma2,delta,delta_t,gamma] x N
    params_out[0 * NB + n] = dt;
    params_out[1 * NB + n] = dx;
    params_out[2 * NB + n] = dy;
    params_out[3 * NB + n] = sigma2;
    params_out[4 * NB + n] = delta;
    params_out[5 * NB + n] = delta_t;
    params_out[6 * NB + n] = gamma;
  }
  __syncthreads();
  // zero-fill padded FxT
  for (int i = tid; i < NB * WP * QP; i += 256) FxT[i] = 0.0f;
  __syncthreads();
  // Fx rows (normalized), stored transposed+padded: FxT[n][w][q]
  for (int i = tid; i < NB * OUTW; i += 256) {
    int n = i / OUTW, q = i % OUTW;
    float mu  = prm[n][1] + ((float)q - 10.0f) * prm[n][4];
    float inv2s2 = 1.0f / (2.0f * prm[n][3]);
    float s = 0.0f;
    for (int w = 0; w < WW; ++w) {
      float d = (float)w - mu;
      s += __expf(-d * d * inv2s2);
    }
    float inv = 1.0f / fmaxf(s, 1e-8f);
    for (int w = 0; w < WW; ++w) {
      float d = (float)w - mu;
      FxT[(n * WP + w) * QP + q] = __expf(-d * d * inv2s2) * inv;
    }
  }
  // Fy rows (normalized): Fy[n][p][h]
  for (int i = tid; i < NB * OUTH; i += 256) {
    int n = i / OUTH, p = i % OUTH;
    float mu  = prm[n][2] + ((float)p - 8.0f) * prm[n][4];
    float inv2s2 = 1.0f / (2.0f * prm[n][3]);
    float s = 0.0f;
    for (int h = 0; h < HH; ++h) {
      float d = (float)h - mu;
      s += __expf(-d * d * inv2s2);
    }
    float inv = 1.0f / fmaxf(s, 1e-8f);
    for (int h = 0; h < HH; ++h) {
      float d = (float)h - mu;
      Fy[(n * OUTH + p) * HH + h] = __expf(-d * d * inv2s2) * inv;
    }
  }
  // temporal lerp tables
  for (int i = tid; i < NB * TT; i += 256) {
    int n = i / TT, t = i % TT;
    float mu = prm[n][0] + ((float)t - 15.0f) * prm[n][5];
    float f0 = floorf(mu);
    float fr = mu - f0;
    int i0 = (int)f0;
    int i1 = i0 + 1;
    tw0[i] = (i0 >= 0 && i0 < TT) ? (1.0f - fr) : 0.0f;
    tw1[i] = (i1 >= 0 && i1 < TT) ? fr : 0.0f;
    ti0[i] = min(max(i0, 0), TT - 1);
    ti1[i] = min(max(i1, 0), TT - 1);
  }
}

// ---------------------------------------------------------------------------
// Stage 1: g[n,c,t] = Fy(16x64) * x(64x44) * FxT(44x20)   via f32 WMMA
// grid = N*C blocks, 128 threads (4 waves), loop over t; x frames streamed
// with double-buffered GLOBAL_LOAD_ASYNC_TO_LDS_B128 (ASYNCcnt).
// One frame = 64 rows x 11 B128 chunks = 704 chunks; LDS row stride 48 floats.
// ---------------------------------------------------------------------------
__device__ __forceinline__ void issue_frame_async(const float* gsrc, float* lbuf,
                                                  int tid) {
  for (int chunk = tid; chunk < HH * 11; chunk += 128) {
    const int h  = chunk / 11;
    const int cw = chunk % 11;
    const float* gp = gsrc + h * WW + cw * 4;
    float*       lp = lbuf + h * WP + cw * 4;
    __builtin_amdgcn_global_load_async_to_lds_b128(
        (as1_v4i*)(uintptr_t)gp, (as3_v4i*)(uintptr_t)lp, 0, 0);
  }
}

__global__ void __launch_bounds__(128)
stage1_kernel(const float* __restrict__ x, const float* __restrict__ ws,
              float* __restrict__ g)
{
  __shared__ float sFy[OUTH * HH];     // 16x64
  __shared__ float sFxT[WP * QP];      // 48x32 (zero padded)
  __shared__ float sX[2][HH * WP];     // double-buffered 64x48 (pad cols zeroed once)
  __shared__ float sT2[OUTH * WP];     // 16x48

  const float* FxT = ws + OFF_FXT;
  const float* Fy  = ws + OFF_FY;

  const int nc   = blockIdx.x;
  const int n    = nc >> 6;
  const int c    = nc & 63;
  const int tid  = threadIdx.x;
  const int lane = tid & 31;
  const int wave = tid >> 5;

  for (int i = tid; i < OUTH * HH; i += 128) sFy[i] = Fy[n * (OUTH * HH) + i];
  for (int i = tid; i < WP * QP; i += 128) sFxT[i] = FxT[n * (WP * QP) + i];
  // zero the pad columns (w=44..47) of both buffers, once
  for (int i = tid; i < HH * 4 * 2; i += 128) {
    int b = i / (HH * 4);
    int r = i % (HH * 4);
    sX[b][(r >> 2) * WP + WW + (r & 3)] = 0.0f;
  }

  const float* xbase = x + ((size_t)(n * CC + c) * TT) * (HH * WW);
  issue_frame_async(xbase, sX[0], tid);   // prime t=0

  const int mrow  = lane & 15;          // M index (or N column within tile)
  const int khalf = (lane >> 4) * 2;    // 0 for lanes 0-15, 2 for lanes 16-31
  const int mbase = (lane < 16) ? 0 : 8;

  for (int t = 0; t < TT; ++t) {
    float* cur = sX[t & 1];
    asm volatile("s_wait_asynccnt 0x0" ::: "memory");
    __syncthreads();   // current buffer ready & previous reads of next buffer done
    if (t + 1 < TT)
      issue_frame_async(xbase + (size_t)(t + 1) * (HH * WW), sX[(t + 1) & 1], tid);

    // Phase A: t2 = Fy * x   (M=16, N=48 -> 3 tiles, K=64)
    if (wave < 3) {
      const int wb = wave * 16;
      v8f acc = {};
      for (int k0 = 0; k0 < HH; k0 += 4) {
        v2f a, b;
        a.x = sFy[mrow * HH + k0 + khalf];
        a.y = sFy[mrow * HH + k0 + khalf + 1];
        b.x = cur[(k0 + khalf) * WP + wb + mrow];
        b.y = cur[(k0 + khalf + 1) * WP + wb + mrow];
        acc = wmma_f32(a, b, acc);
      }
#pragma unroll
      for (int r = 0; r < 8; ++r)
        sT2[(mbase + r) * WP + wb + mrow] = acc[r];
    }
    __syncthreads();

    // Phase B: g = t2 * FxT   (M=16, N=32 -> 2 tiles, K=48)
    if (wave < 2) {
      const int qb = wave * 16;
      v8f acc = {};
      for (int k0 = 0; k0 < WP; k0 += 4) {
        v2f a, b;
        a.x = sT2[mrow * WP + k0 + khalf];
        a.y = sT2[mrow * WP + k0 + khalf + 1];
        b.x = sFxT[(k0 + khalf) * QP + qb + mrow];
        b.y = sFxT[(k0 + khalf + 1) * QP + qb + mrow];
        acc = wmma_f32(a, b, acc);
      }
      const int q = qb + mrow;
      if (q < OUTW) {
        float* gp = g + (((size_t)(n * CC + c) * TT + t) * HWSZ);
#pragma unroll
        for (int r = 0; r < 8; ++r)
          gp[(mbase + r) * OUTW + q] = acc[r];
      }
    }
  }
}

// ---------------------------------------------------------------------------
// Stage 2: out[n,o,t',h,w] = bf[o] + sum_c Wf[o,c]*gamma*(w0*g[..i0..]+w1*g[..i1..])
// grid = N*T blocks, 256 threads (8 waves). GEMM 64x64x320 via WMMA, hw split
// into two halves of 160 to keep static LDS < 64 KB. g is L2-resident.
// ---------------------------------------------------------------------------
__global__ void __launch_bounds__(256)
stage2_kernel(const float* __restrict__ g, const float* __restrict__ Wf,
              const float* __restrict__ bf, const float* __restrict__ ws,
              float* __restrict__ out)
{
  __shared__ float sWf[CC * CC];      // 64x64  (16 KB)
  __shared__ float sS[CC * 160];      // 64x160 (40 KB)
  __shared__ float sbf[CC];

  const float* gam = ws + OFF_GAM;
  const float* tw0 = ws + OFF_TW0;
  const float* tw1 = ws + OFF_TW1;
  const int*   ti0 = (const int*)(ws + OFF_TI0);
  const int*   ti1 = (const int*)(ws + OFF_TI1);

  const int ntt = blockIdx.x;
  const int n   = ntt / TT;
  const int tt  = ntt % TT;
  const int tid  = threadIdx.x;
  const int lane = tid & 31;
  const int wave = tid >> 5;

  for (int i = tid; i < CC * CC; i += 256) sWf[i] = Wf[i];
  if (tid < CC) sbf[tid] = bf[tid];

  const float gmm = gam[n];
  const float w0  = tw0[n * TT + tt] * gmm;
  const float w1  = tw1[n * TT + tt] * gmm;
  const int   i0  = ti0[n * TT + tt];
  const int   i1  = ti1[n * TT + tt];
  const float* g0 = g + (((size_t)n * CC * TT + i0) * HWSZ);
  const float* g1 = g + (((size_t)n * CC * TT + i1) * HWSZ);

  const int mrow  = lane & 15;
  const int khalf = (lane >> 4) * 2;
  const int mbase = (lane < 16) ? 0 : 8;

  float* outp = out + ((size_t)n * CC * TT + tt) * HWSZ;  // + o*TT*HWSZ + hw

  for (int half = 0; half < 2; ++half) {
    const int hw0 = half * 160;
    __syncthreads();
    // blend two frames * gamma into LDS: sS[c][hwl]
    for (int i = tid; i < CC * 160; i += 256) {
      int c = i / 160, hwl = i % 160;
      size_t off = (size_t)c * TT * HWSZ + hw0 + hwl;
      sS[i] = w0 * g0[off] + w1 * g1[off];
    }
    __syncthreads();
    // GEMM: 4 M-tiles x 10 N-tiles, K=64 -> 40 tiles over 8 waves
    for (int tile = wave; tile < 40; tile += 8) {
      const int mt = tile / 10;
      const int nt = tile % 10;
      const int ob = mt * 16;
      const int hb = nt * 16;
      v8f acc = {};
      for (int k0 = 0; k0 < CC; k0 += 4) {
        v2f a, b;
        a.x = sWf[(ob + mrow) * CC + k0 + khalf];
        a.y = sWf[(ob + mrow) * CC + k0 + khalf + 1];
        b.x = sS[(k0 + khalf) * 160 + hb + mrow];
        b.y = sS[(k0 + khalf + 1) * 160 + hb + mrow];
        acc = wmma_f32(a, b, acc);
      }
      const int hw = hw0 + hb + mrow;
#pragma unroll
      for (int r = 0; r < 8; ++r) {
        const int o = ob + mbase + r;
        outp[(size_t)o * TT * HWSZ + hw] = acc[r] + sbf[o];
      }
    }
  }
}

// ---------------------------------------------------------------------------
extern "C" void kernel_launch(void* const* d_in, const int* in_sizes, int n_in,
                              void* d_out, int out_size, void* d_ws, size_t ws_size,
                              hipStream_t stream) {
  const float* x       = (const float*)d_in[0];
  const float* param_x = (const float*)d_in[1];
  const float* W1      = (const float*)d_in[2];
  const float* b1      = (const float*)d_in[3];
  const float* W2      = (const float*)d_in[4];
  const float* b2      = (const float*)d_in[5];
  const float* Wf      = (const float*)d_in[6];
  const float* bf      = (const float*)d_in[7];
  float* out = (float*)d_out;
  float* ws  = (float*)d_ws;
  float* gbuf = ws + OFF_G;

  prep_kernel<<<1, 256, 0, stream>>>(param_x, W1, b1, W2, b2, ws, out + OUT_MAIN);
  stage1_kernel<<<NB * CC, 128, 0, stream>>>(x, ws, gbuf);
  stage2_kernel<<<NB * TT, 256, 0, stream>>>(gbuf, Wf, bf, ws, out);
}